// TerritoryGNN_3015067041913
// MI455X (gfx1250) — compile-verified
//
#include <hip/hip_runtime.h>

typedef float v2f __attribute__((ext_vector_type(2)));
typedef float v8f __attribute__((ext_vector_type(8)));

#define C_DIM 128
// LDS layout for W: one "t-row" per K-pair (t = k/2) holding 128 interleaved
// (W[2t][c], W[2t+1][c]) pairs = 256 dwords, padded to 288 dwords so that
// consecutive t-rows are skewed by 32 banks (288 mod 64 == 32). The hi
// half-wave (lanes 16-31) reads t+1 while lanes 0-15 read t, landing on
// disjoint bank sets. 64 t-rows * 288 dwords * 4B = 72 KB.
#define LDS_TROW 288

// ---------------- degree / normalization ----------------

__global__ void init_deg_kernel(float* deg, int n) {
    int i = blockIdx.x * blockDim.x + threadIdx.x;
    if (i < n) deg[i] = 1.0f;  // self-loop contribution
}

__global__ void edge_deg_kernel(const int* __restrict__ dst, float* deg, int e) {
    int i = blockIdx.x * blockDim.x + threadIdx.x;
    if (i < e) atomicAdd(&deg[dst[i]], 1.0f);
}

__global__ void dinv_kernel(const float* __restrict__ deg, float* dinv, int n) {
    int i = blockIdx.x * blockDim.x + threadIdx.x;
    if (i < n) {
        float d = deg[i];
        dinv[i] = d > 0.0f ? rsqrtf(d) : 0.0f;
    }
}

// ---------------- GEMM: H = X @ W  (fp32 WMMA 16x16x4) ----------------
// Block = 256 threads = 8 waves. W staged in LDS, K-pair swizzled (see above)
// so each lane's B fragment is a single aligned ds_load_b64. Each wave owns
// one 16-row M-tile and accumulates all 8 N-tiles (8 x v8f), reusing its A
// fragment across the N dimension.

__global__ void __launch_bounds__(256) gemm_wmma_f32(const float* __restrict__ X,
                                                     const float* __restrict__ W,
                                                     float* __restrict__ H, int nrows) {
    extern __shared__ float sW[];  // 64 * LDS_TROW dwords
    int tid = threadIdx.x;
    for (int i = tid; i < C_DIM * C_DIM; i += blockDim.x) {
        int r = i >> 7;   // k index
        int c = i & 127;  // column
        sW[(r >> 1) * LDS_TROW + (c << 1) + (r & 1)] = W[i];
    }
    __syncthreads();

    int wave   = tid >> 5;
    int lane   = tid & 31;
    int tile   = blockIdx.x * (blockDim.x >> 5) + wave;
    int ntiles = nrows >> 4;
    if (tile >= ntiles) return;

    int row0 = tile << 4;
    int lrow = lane & 15;
    int hi   = lane >> 4;  // 0 for lanes 0-15, 1 for lanes 16-31

    // A fragment: lane L<16 holds row L, K = {k, k+1}; L>=16 holds K = {k+2, k+3}
    const float2* xrow = (const float2*)(X + (size_t)(row0 + lrow) * C_DIM);

    v8f acc[8] = {};

    for (int k = 0; k < C_DIM; k += 4) {
        int    t  = (k >> 1) + hi;  // K-pair index this half-wave consumes
        float2 af = xrow[t];
        v2f    a;
        a.x = af.x;
        a.y = af.y;
        const float2* brow = (const float2*)(sW + t * LDS_TROW);
#pragma unroll
        for (int n = 0; n < 8; ++n) {
            int    col = (n << 4) + lrow;
            float2 bf  = brow[col];  // (W[kb][col], W[kb+1][col]) -> one ds_load_b64
            v2f    b;
            b.x = bf.x;
            b.y = bf.y;
            // (neg_a, A, neg_b, B, c_mod, C, reuse_a, reuse_b)
            acc[n] = __builtin_amdgcn_wmma_f32_16x16x4_f32(
                false, a, false, b, (short)0, acc[n], false, false);
        }
    }

    // C/D layout: VGPR i -> row (i) for lanes 0-15, row (i+8) for lanes 16-31; col = lane%16
#pragma unroll
    for (int i = 0; i < 8; ++i) {
        int    r    = row0 + i + (hi << 3);
        float* orow = H + (size_t)r * C_DIM;
#pragma unroll
        for (int n = 0; n < 8; ++n) {
            orow[(n << 4) + lrow] = acc[n][i];
        }
    }
}

// ---------------- aggregation ----------------
// Initialize accumulator with the self-loop term (also serves as zero-init,
// since the harness does not re-zero buffers between replays).
__global__ void agg_self_kernel(const float* __restrict__ h,
                                const float* __restrict__ dinv,
                                float* __restrict__ out, int n) {
    int idx   = blockIdx.x * blockDim.x + threadIdx.x;  // over n * 32 float4s
    int total = n * (C_DIM / 4);
    if (idx < total) {
        int   row = idx >> 5;  // C_DIM/4 == 32
        float s   = dinv[row];
        s *= s;
        float4 v = ((const float4*)h)[idx];
        float4 o;
        o.x = v.x * s; o.y = v.y * s; o.z = v.z * s; o.w = v.w * s;
        ((float4*)out)[idx] = o;
    }
}

// One wave per edge: 32 lanes x float4 = 128 channels. Gather h[src], scale,
// scatter-add into out[dst] with global_atomic_add_f32. With 192 MB L2 the
// 51 MB feature matrix is L2-resident, so the random gathers mostly hit L2.
__global__ void agg_edge_kernel(const float* __restrict__ h,
                                const float* __restrict__ dinv,
                                const int* __restrict__ src,
                                const int* __restrict__ dst,
                                float* __restrict__ out, int e) {
    int gwave  = (blockIdx.x * blockDim.x + threadIdx.x) >> 5;
    int lane   = threadIdx.x & 31;
    int nwaves = (gridDim.x * blockDim.x) >> 5;
    for (int ed = gwave; ed < e; ed += nwaves) {
        int   s  = src[ed];
        int   d  = dst[ed];
        float nm = dinv[s] * dinv[d];
        float4 v = ((const float4*)(h + (size_t)s * C_DIM))[lane];
        float* od = out + (size_t)d * C_DIM + lane * 4;
        atomicAdd(od + 0, v.x * nm);
        atomicAdd(od + 1, v.y * nm);
        atomicAdd(od + 2, v.z * nm);
        atomicAdd(od + 3, v.w * nm);
    }
}

__global__ void bias_relu_kernel(const float* __restrict__ in,
                                 const float* __restrict__ b,
                                 float* __restrict__ out, int n) {
    int idx   = blockIdx.x * blockDim.x + threadIdx.x;
    int total = n * C_DIM;
    if (idx < total) {
        float v = in[idx] + b[idx & (C_DIM - 1)];
        out[idx] = v > 0.0f ? v : 0.0f;
    }
}

// ---------------- host launch ----------------

extern "C" void kernel_launch(void* const* d_in, const int* in_sizes, int n_in,
                              void* d_out, int out_size, void* d_ws, size_t ws_size,
                              hipStream_t stream) {
    (void)n_in; (void)out_size; (void)ws_size;
    const float* x  = (const float*)d_in[0];
    const int*   ei = (const int*)d_in[1];
    const float* W1 = (const float*)d_in[2];
    const float* b1 = (const float*)d_in[3];
    const float* W2 = (const float*)d_in[4];
    const float* b2 = (const float*)d_in[5];

    int n = in_sizes[0] / C_DIM;  // 100000
    int e = in_sizes[1] / 2;      // 1600000
    const int* src = ei;          // edge_index row 0
    const int* dst = ei + e;      // edge_index row 1

    // workspace layout
    char*  ws   = (char*)d_ws;
    float* deg  = (float*)ws;
    float* dinv = deg + n;
    size_t hdr  = ((size_t)2 * n * sizeof(float) + 255) & ~(size_t)255;
    float* buf0 = (float*)(ws + hdr);        // H (post-GEMM features)
    float* buf1 = buf0 + (size_t)n * C_DIM;  // aggregation accumulator / layer-1 output
    float* out  = (float*)d_out;

    const int T = 256;

    // ---- symmetric normalization (graph is shared by both layers) ----
    init_deg_kernel<<<(n + T - 1) / T, T, 0, stream>>>(deg, n);
    edge_deg_kernel<<<(e + T - 1) / T, T, 0, stream>>>(dst, deg, e);
    dinv_kernel<<<(n + T - 1) / T, T, 0, stream>>>(deg, dinv, n);

    int    ntiles  = n >> 4;            // 6250 M-tiles
    int    gblocks = (ntiles + 7) / 8;  // 8 waves per block
    int    selfBlk = (n * (C_DIM / 4) + T - 1) / T;
    int    elemBlk = (n * C_DIM + T - 1) / T;
    size_t ldsSz   = (size_t)(C_DIM / 2) * LDS_TROW * sizeof(float);  // 72 KB

    // ---- layer 1 ----
    gemm_wmma_f32<<<gblocks, 256, ldsSz, stream>>>(x, W1, buf0, n);
    agg_self_kernel<<<selfBlk, T, 0, stream>>>(buf0, dinv, buf1, n);
    agg_edge_kernel<<<2048, 256, 0, stream>>>(buf0, dinv, src, dst, buf1, e);
    bias_relu_kernel<<<elemBlk, T, 0, stream>>>(buf1, b1, buf1, n);

    // ---- layer 2 (aggregate directly into d_out) ----
    gemm_wmma_f32<<<gblocks, 256, ldsSz, stream>>>(buf1, W2, buf0, n);
    agg_self_kernel<<<selfBlk, T, 0, stream>>>(buf0, dinv, out, n);
    agg_edge_kernel<<<2048, 256, 0, stream>>>(buf0, dinv, src, dst, out, e);
    bias_relu_kernel<<<elemBlk, T, 0, stream>>>(out, b2, out, n);
}